// MultimodalEnhancedHyperModel_49185965473982
// MI455X (gfx1250) — compile-verified
//
#include <hip/hip_runtime.h>
#include <math.h>

#define EPS 1e-8f

typedef __attribute__((ext_vector_type(2))) float v2f;
typedef __attribute__((ext_vector_type(8))) float v8f;

// ---------------------------------------------------------------------------
// fp32 WMMA GEMM: C[M,N] = A[M,K] @ B[K,Np] (+ bias[N])
// B must be column-padded so Np = 16*tilesN (no load guards needed).
// One wave32 computes a 64x16 C block: 4 stacked 16x16 tiles sharing each
// B fragment (V_WMMA_F32_16X16X4_F32, K step 4).
//   A-frag: lane L holds A[m + L%16][k + 2*(L/16) + {0,1}]   (v2f)
//   B-frag: lane L holds B[k + 2*(L/16) + {0,1}][n0 + L%16]  (v2f)
//   C-frag: VGPR j -> row m + j + 8*(L/16), col n0 + L%16
// M multiple of 64, K multiple of 4 (true for all calls here).
// ---------------------------------------------------------------------------
__global__ __launch_bounds__(256) void wmma_gemm_f32(
    const float* __restrict__ A, int lda,
    const float* __restrict__ Bp, int ldb,
    float* __restrict__ C, int ldc,
    const float* __restrict__ bias,
    int M, int N, int K, int tilesN)
{
  const int wave = threadIdx.x >> 5;
  const int lane = threadIdx.x & 31;
  const int tile = blockIdx.x * 8 + wave;
  const int rowGroups = M >> 6;                 // 64 rows per group
  if (tile >= rowGroups * tilesN) return;

  const int tg = tile / tilesN;
  const int tn = tile - tg * tilesN;
  const int m0 = tg << 6;
  const int n0 = tn << 4;
  const int r    = lane & 15;
  const int half = lane >> 4;
  const int col  = n0 + r;

  v8f acc0 = {}, acc1 = {}, acc2 = {}, acc3 = {};
  const float* aptr = A + (size_t)(m0 + r) * lda + 2 * half;
  const size_t astep = (size_t)16 * lda;
  const float* bptr = Bp + (size_t)(2 * half) * ldb + col;

  for (int k = 0; k < K; k += 4) {
    v2f b;
    b.x = bptr[0];                              // B[k+2h  ][col]
    b.y = bptr[ldb];                            // B[k+2h+1][col]
    v2f a0 = *(const v2f*)(aptr);
    v2f a1 = *(const v2f*)(aptr + astep);
    v2f a2 = *(const v2f*)(aptr + 2 * astep);
    v2f a3 = *(const v2f*)(aptr + 3 * astep);
    acc0 = __builtin_amdgcn_wmma_f32_16x16x4_f32(false, a0, false, b, (short)0, acc0, false, false);
    acc1 = __builtin_amdgcn_wmma_f32_16x16x4_f32(false, a1, false, b, (short)0, acc1, false, false);
    acc2 = __builtin_amdgcn_wmma_f32_16x16x4_f32(false, a2, false, b, (short)0, acc2, false, false);
    acc3 = __builtin_amdgcn_wmma_f32_16x16x4_f32(false, a3, false, b, (short)0, acc3, false, false);
    aptr += 4;
    bptr += (size_t)4 * ldb;
  }

  if (col < N) {                                // store guard only (outside loop)
    float bb = bias ? bias[col] : 0.0f;
#pragma unroll
    for (int j = 0; j < 8; ++j) {
      int rr = j + 8 * half;
      C[(size_t)(m0 + rr)      * ldc + col] = acc0[j] + bb;
      C[(size_t)(m0 + 16 + rr) * ldc + col] = acc1[j] + bb;
      C[(size_t)(m0 + 32 + rr) * ldc + col] = acc2[j] + bb;
      C[(size_t)(m0 + 48 + rr) * ldc + col] = acc3[j] + bb;
    }
  }
}

// Zero-pad columns: src[K,N] -> dst[K,Np]
__global__ void pad_cols(const float* __restrict__ src, float* __restrict__ dst,
                         int K, int N, int Np) {
  int i = blockIdx.x * blockDim.x + threadIdx.x;
  if (i >= K * Np) return;
  int k = i / Np;
  int c = i - k * Np;
  dst[i] = (c < N) ? src[(size_t)k * N + c] : 0.0f;
}

// ---------------------------------------------------------------------------
// Utility / sparse-hypergraph kernels
// ---------------------------------------------------------------------------
__global__ void zero_f32(float* p, int n) {
  int i = blockIdx.x * blockDim.x + threadIdx.x;
  if (i < n) p[i] = 0.0f;
}

// Column means of x [N,D] -> colmean[D]; one block per column.
__global__ __launch_bounds__(256) void col_mean(const float* __restrict__ x,
                                                float* __restrict__ colmean,
                                                int N, int D) {
  __shared__ float sm[256];
  int c = blockIdx.x;
  float s = 0.0f;
  for (int r = threadIdx.x; r < N; r += blockDim.x) s += x[(size_t)r * D + c];
  sm[threadIdx.x] = s;
  __syncthreads();
  for (int o = 128; o > 0; o >>= 1) {
    if (threadIdx.x < o) sm[threadIdx.x] += sm[threadIdx.x + o];
    __syncthreads();
  }
  if (threadIdx.x == 0) colmean[c] = sm[0] / (float)N;
}

__global__ __launch_bounds__(256) void alpha_k(const float* __restrict__ colmean,
                                               const float* __restrict__ gate_w,
                                               const float* __restrict__ gate_b,
                                               float* __restrict__ alpha, int D) {
  __shared__ float sm[256];
  int t = threadIdx.x;
  float s = (t < D) ? colmean[t] * gate_w[t] : 0.0f;
  sm[t] = s;
  __syncthreads();
  for (int o = 128; o > 0; o >>= 1) {
    if (t < o) sm[t] += sm[t + o];
    __syncthreads();
  }
  if (t == 0) alpha[0] = 1.0f / (1.0f + __expf(-(sm[0] + gate_b[0])));
}

// Per-edge dedup (matches JAX .set semantics), node-degree counts, edge degree.
__global__ void edge_deg(const int* __restrict__ en, float* __restrict__ DvCnt,
                         float* __restrict__ de, unsigned* __restrict__ mask,
                         int E, int DEG) {
  int e = blockIdx.x * blockDim.x + threadIdx.x;
  if (e >= E) return;
  const int* idx = en + (size_t)e * DEG;
  unsigned m = 0;
  int distinct = 0;
  for (int d = 0; d < DEG; d++) {
    int n = idx[d];
    bool dup = false;
    for (int d2 = 0; d2 < d; d2++)
      if (idx[d2] == n) { dup = true; break; }
    if (!dup) {
      m |= (1u << d);
      distinct++;
      atomicAdd(&DvCnt[n], 1.0f);
    }
  }
  mask[e] = m;
  de[e] = 1.0f / ((float)distinct * (1.0f / (float)DEG) + EPS);
}

// DvCnt -> dv = 1/sqrt(cnt/DEG + eps), in place.
__global__ void dv_k(float* Dv, int N, float invdeg) {
  int i = blockIdx.x * blockDim.x + threadIdx.x;
  if (i < N) Dv[i] = rsqrtf(Dv[i] * invdeg + EPS);
}

// tmp[e,:] = de[e]*(1/DEG) * sum_{distinct n in e} dv[n]*out1[n,:]
__global__ __launch_bounds__(256) void edge_gather(
    const int* __restrict__ en, const unsigned* __restrict__ mask,
    const float* __restrict__ dv, const float* __restrict__ de,
    const float* __restrict__ out1, float* __restrict__ tmp, int D, int DEG) {
  int e = blockIdx.x;
  int c = threadIdx.x;
  if (c >= D) return;
  unsigned m = mask[e];
  const int* idx = en + (size_t)e * DEG;
  float acc = 0.0f;
  for (int s = 0; s < DEG; s++) {
    if (m & (1u << s)) {
      int n = idx[s];
      acc += dv[n] * out1[(size_t)n * D + c];
    }
  }
  tmp[(size_t)e * D + c] = acc * de[e] * (1.0f / (float)DEG);
}

// hyper[n,:] += dv[n]*(1/DEG) * tmp[e,:] for each distinct n in e.
__global__ __launch_bounds__(256) void edge_scatter(
    const int* __restrict__ en, const unsigned* __restrict__ mask,
    const float* __restrict__ dv, const float* __restrict__ tmp,
    float* __restrict__ hyper, int D, int DEG) {
  int e = blockIdx.x;
  int c = threadIdx.x;
  if (c >= D) return;
  unsigned m = mask[e];
  const int* idx = en + (size_t)e * DEG;
  float tv = tmp[(size_t)e * D + c];
  for (int s = 0; s < DEG; s++) {
    if (m & (1u << s)) {
      int n = idx[s];
      atomicAdd(&hyper[(size_t)n * D + c], dv[n] * (1.0f / (float)DEG) * tv);
    }
  }
}

// gnn_feat = alpha*out1 + (1-alpha)*hyper + b[col]
__global__ void gnn_feat_k(const float* __restrict__ out1,
                           const float* __restrict__ hyper,
                           const float* __restrict__ bvec,
                           const float* __restrict__ alpha,
                           float* __restrict__ gf, int total, int D) {
  int i = blockIdx.x * blockDim.x + threadIdx.x;
  if (i >= total) return;
  float a = alpha[0];
  gf[i] = a * out1[i] + (1.0f - a) * hyper[i] + bvec[i % D];
}

// text -> tp -> v -> att2 -> cgate  (per-graph constants; kv_len==1 => attn==1)
__global__ __launch_bounds__(128) void text_chain(
    const float* __restrict__ text, const float* __restrict__ text_w,
    const float* __restrict__ text_b, const float* __restrict__ in_w,
    const float* __restrict__ in_b, const float* __restrict__ aw,
    const float* __restrict__ ab, const float* __restrict__ fgate_w,
    const float* __restrict__ fgate_b, float* __restrict__ att2,
    float* __restrict__ cg, int D, int H) {
  int b = blockIdx.x;
  int h = threadIdx.x;
  __shared__ float s0[128], s1[128];
  float acc = text_b[h];
  for (int d = 0; d < D; d++) acc += text[(size_t)b * D + d] * text_w[(size_t)d * H + h];
  s0[h] = acc;
  __syncthreads();
  acc = in_b[2 * H + h];
  for (int j = 0; j < H; j++) acc += s0[j] * in_w[(size_t)j * 3 * H + 2 * H + h];
  s1[h] = acc;
  __syncthreads();
  acc = ab[h];
  for (int j = 0; j < H; j++) acc += s1[j] * aw[(size_t)j * H + h];
  att2[(size_t)b * H + h] = acc;
  s0[h] = acc;
  __syncthreads();
  acc = fgate_b[h];
  for (int j = 0; j < H; j++) acc += s0[j] * fgate_w[(size_t)(H + j) * H + h];
  cg[(size_t)b * H + h] = acc;
}

// g = sigmoid(t1 + cg[b]); fused = g*gp + (1-g)*att2[b]
__global__ void fuse_k(const float* __restrict__ t1, const float* __restrict__ cg,
                       const float* __restrict__ att2, const float* __restrict__ gp,
                       float* __restrict__ fused, int total, int H, int Nn) {
  int i = blockIdx.x * blockDim.x + threadIdx.x;
  if (i >= total) return;
  int n = i / H;
  int h = i - n * H;
  int b = n / Nn;
  float g = 1.0f / (1.0f + __expf(-(t1[i] + cg[(size_t)b * H + h])));
  fused[i] = g * gp[i] + (1.0f - g) * att2[(size_t)b * H + h];
}

// ---------------------------------------------------------------------------
// Host-side launch
// ---------------------------------------------------------------------------
static inline size_t align256(size_t x) { return (x + 255) & ~(size_t)255; }

extern "C" void kernel_launch(void* const* d_in, const int* in_sizes, int n_in,
                              void* d_out, int out_size, void* d_ws, size_t ws_size,
                              hipStream_t stream) {
  (void)in_sizes; (void)n_in; (void)out_size; (void)ws_size;
  const int N = 8192, D = 200, Dp = 208, Bq = 4, E = 4096, DEG = 16, H = 128, Nn = 2048;

  const float* x       = (const float*)d_in[0];
  const float* text    = (const float*)d_in[1];
  const float* W       = (const float*)d_in[2];
  const float* bvec    = (const float*)d_in[3];
  const float* gate_w  = (const float*)d_in[4];
  const float* gate_b  = (const float*)d_in[5];
  const float* gnn_w   = (const float*)d_in[6];
  const float* gnn_b   = (const float*)d_in[7];
  const float* text_w  = (const float*)d_in[8];
  const float* text_b  = (const float*)d_in[9];
  const float* in_w    = (const float*)d_in[10];
  const float* in_b    = (const float*)d_in[11];
  const float* aw      = (const float*)d_in[12];
  const float* ab      = (const float*)d_in[13];
  const float* fgate_w = (const float*)d_in[14];
  const float* fgate_b = (const float*)d_in[15];
  const float* outp_w  = (const float*)d_in[16];
  const float* outp_b  = (const float*)d_in[17];
  const int*   en      = (const int*)d_in[18];
  float* outp = (float*)d_out;

  // ---- workspace carve-up ----
  char* base = (char*)d_ws;
  size_t off = 0;
  auto alloc = [&](size_t floats) {
    float* p = (float*)(base + off);
    off = align256(off + floats * sizeof(float));
    return p;
  };
  float*    out1   = alloc((size_t)N * D);    // x @ W
  float*    hyper  = alloc((size_t)N * D);    // Laplacian-propagated features
  float*    gf     = alloc((size_t)N * D);    // gnn_feat
  float*    gp     = alloc((size_t)N * H);
  float*    t1     = alloc((size_t)N * H);    // gp @ fgate_w[:H]
  float*    fused  = alloc((size_t)N * H);
  float*    Wp     = alloc((size_t)D * Dp);   // W padded to 208 cols
  float*    outwp  = alloc((size_t)H * Dp);   // outp_w padded to 208 cols
  float*    dv     = alloc(N);                // counts, then 1/sqrt(Dv+eps)
  float*    de     = alloc(E);
  unsigned* mask   = (unsigned*)alloc(E);
  float*    etmp   = alloc((size_t)E * D);
  float*    cmean  = alloc(D);
  float*    alpha  = alloc(1);
  float*    att2   = alloc((size_t)Bq * H);
  float*    cg     = alloc((size_t)Bq * H);

  auto gemm = [&](const float* A, int lda, const float* B, int ldb, float* C,
                  int ldc, const float* bias, int M, int Nc, int K) {
    int tilesN = (Nc + 15) / 16;
    int tiles = (M >> 6) * tilesN;              // 64-row groups per wave
    int blocks = (tiles + 7) / 8;
    wmma_gemm_f32<<<blocks, 256, 0, stream>>>(A, lda, B, ldb, C, ldc, bias,
                                              M, Nc, K, tilesN);
  };

  // 0) zero accumulators; pad the two N=200 weight matrices to 208 cols
  zero_f32<<<(N + 255) / 256, 256, 0, stream>>>(dv, N);
  zero_f32<<<((N * D) + 255) / 256, 256, 0, stream>>>(hyper, N * D);
  pad_cols<<<((D * Dp) + 255) / 256, 256, 0, stream>>>(W, Wp, D, D, Dp);
  pad_cols<<<((H * Dp) + 255) / 256, 256, 0, stream>>>(outp_w, outwp, H, D, Dp);

  // 1) out1 = x @ W                      [8192,200]x[200,208p]
  gemm(x, D, Wp, Dp, out1, D, nullptr, N, D, D);

  // 2) alpha = sigmoid(mean(x,0) @ gate_w + gate_b)
  col_mean<<<D, 256, 0, stream>>>(x, cmean, N, D);
  alpha_k<<<1, 256, 0, stream>>>(cmean, gate_w, gate_b, alpha, D);

  // 3) hypergraph: degrees -> edge gather -> node scatter
  edge_deg<<<(E + 255) / 256, 256, 0, stream>>>(en, dv, de, mask, E, DEG);
  dv_k<<<(N + 255) / 256, 256, 0, stream>>>(dv, N, 1.0f / (float)DEG);
  edge_gather<<<E, 256, 0, stream>>>(en, mask, dv, de, out1, etmp, D, DEG);
  edge_scatter<<<E, 256, 0, stream>>>(en, mask, dv, etmp, hyper, D, DEG);

  // 4) gnn_feat = alpha*out1 + (1-alpha)*hyper + b
  gnn_feat_k<<<((N * D) + 255) / 256, 256, 0, stream>>>(out1, hyper, bvec,
                                                        alpha, gf, N * D, D);

  // 5) gp = gnn_feat @ gnn_w + gnn_b     [8192,200]x[200,128]  (no pad needed)
  gemm(gf, D, gnn_w, H, gp, H, gnn_b, N, H, D);

  // 6) per-graph text chain (attention with kv_len==1 collapses to v-path)
  text_chain<<<Bq, H, 0, stream>>>(text, text_w, text_b, in_w, in_b, aw, ab,
                                   fgate_w, fgate_b, att2, cg, D, H);

  // 7) t1 = gp @ fgate_w[:H,:]           [8192,128]x[128,128]
  gemm(gp, H, fgate_w, H, t1, H, nullptr, N, H, H);

  // 8) gated fusion
  fuse_k<<<((N * H) + 255) / 256, 256, 0, stream>>>(t1, cg, att2, gp, fused,
                                                    N * H, H, Nn);

  // 9) out = fused @ outp_w + outp_b     [8192,128]x[128,208p] -> d_out
  gemm(fused, H, outwp, Dp, outp, D, outp_b, N, D, H);
}